// gcomgraphcombinations_62826781606169
// MI455X (gfx1250) — compile-verified
//
#include <hip/hip_runtime.h>
#include <hip/hip_bf16.h>
#include <stdint.h>

// Problem geometry (from reference): x is (64, 96, 8, 8) f32.
// out[b,i,j,p,q,s] : s=0 -> x[b,i,p,q], s=1 -> x[b,j,p,q]
#define B_DIM        64
#define GS           96
#define TILE_FLOATS  64                 // 8*8 floats = 256 B per tile
#define BATCH_FLOATS (GS * TILE_FLOATS) // 6144 floats = 24 KB per batch slab
#define BLK_FLOATS   128                // one (b,i,j) output block: 8*8*2 floats = 512 B

typedef __attribute__((ext_vector_type(4))) float v4f;
typedef __attribute__((ext_vector_type(2))) float v2f;
typedef __attribute__((ext_vector_type(4))) int   v4i;

// CDNA5 async global->LDS path, feature-gated so the file always compiles.
#if defined(__has_builtin)
#  if __has_builtin(__builtin_amdgcn_global_load_async_to_lds_b128)
#    define HAVE_ASYNC_LDS 1
#  endif
#  if __has_builtin(__builtin_amdgcn_s_wait_asynccnt)
#    define HAVE_WAIT_ASYNC 1
#  endif
#endif

#if defined(HAVE_ASYNC_LDS)
// Param 1 prints as "int4 __device__ *"  -> AS1 (global) int4*
// Param 2 is the LDS destination         -> AS3 (shared) int4*
typedef __attribute__((address_space(1))) v4i glb_v4i;
typedef __attribute__((address_space(3))) v4i lds_v4i;
#endif

__global__ __launch_bounds__(256)
void bcast_pairs_kernel(const float* __restrict__ x, float* __restrict__ out) {
    __shared__ float tiles[BATCH_FLOATS];   // all 96 tiles of batch b (24 KB)

    const int wg  = blockIdx.x;             // wg = b*96 + i
    const int b   = wg / GS;
    const int i   = wg - b * GS;
    const int tid = threadIdx.x;            // 0..255 (8 wave32s)

    const float* src = x + (size_t)b * BATCH_FLOATS;

    // ---- Stage x[b] (24 KB) into LDS: 1536 float4s across 256 threads ----
#if defined(HAVE_ASYNC_LDS)
    {
        // Low 32 bits of a generic LDS pointer == LDS byte offset == AS3 repr.
        const uint32_t lbase = (uint32_t)(uintptr_t)(&tiles[0]);
        #pragma unroll
        for (int k = 0; k < BATCH_FLOATS / 4 / 256; ++k) {
            const int idx = tid + k * 256;  // float4 index
            __builtin_amdgcn_global_load_async_to_lds_b128(
                (glb_v4i*)(uintptr_t)(src + 4 * idx),
                (lds_v4i*)(uint32_t)(lbase + 16u * (uint32_t)idx),
                0, 0);
        }
#if defined(HAVE_WAIT_ASYNC)
        __builtin_amdgcn_s_wait_asynccnt(0);
#else
        asm volatile("s_wait_asynccnt 0" ::: "memory");
#endif
    }
#else
    {
        #pragma unroll
        for (int k = 0; k < BATCH_FLOATS / 4 / 256; ++k) {
            const int idx = tid + k * 256;
            ((v4f*)tiles)[idx] = ((const v4f*)src)[idx];
        }
    }
#endif
    __syncthreads();

    // ---- Emit interleaved pairs ----
    // Wave w handles j = w, w+8, ... ; lane t covers elements 2t, 2t+1 of the
    // 64-element tile, i.e. bytes [16t, 16t+16) of the 512 B (b,i,j) block.
    const int lane = tid & 31;
    const int wave = tid >> 5;

    const v2f xi = *(const v2f*)&tiles[i * TILE_FLOATS + lane * 2];

    float* outb = out + (size_t)wg * (size_t)GS * BLK_FLOATS + (size_t)lane * 4;

    #pragma unroll 4
    for (int j = wave; j < GS; j += 8) {
        const v2f xj = *(const v2f*)&tiles[j * TILE_FLOATS + lane * 2];
        v4f v;
        v.x = xi.x; v.y = xj.x; v.z = xi.y; v.w = xj.y;
        // 302 MB single-use stream (> 192 MB L2): non-temporal store
        __builtin_nontemporal_store(v, (v4f*)(outb + (size_t)j * BLK_FLOATS));
    }
}

extern "C" void kernel_launch(void* const* d_in, const int* in_sizes, int n_in,
                              void* d_out, int out_size, void* d_ws, size_t ws_size,
                              hipStream_t stream) {
    const float* x  = (const float*)d_in[0];
    float*       o  = (float*)d_out;
    (void)in_sizes; (void)n_in; (void)out_size; (void)d_ws; (void)ws_size;

    dim3 grid(B_DIM * GS);   // 6144 workgroups, one per (b, i)
    dim3 block(256);         // 8 wave32s
    hipLaunchKernelGGL(bcast_pairs_kernel, grid, block, 0, stream, x, o);
}